// MXQuantizer_66288525247091
// MI455X (gfx1250) — compile-verified
//
#include <hip/hip_runtime.h>
#include <hip/hip_bf16.h>

typedef float v4f __attribute__((ext_vector_type(4)));

#define WAVES_PER_BLOCK 8
#define DEPTH 4   // async pipeline depth (s_wait_asynccnt literal below must be DEPTH-1)

// xor-swap within 8 consecutive lanes: group-of-32 swizzle, xor_mask=m, and_mask=0x1f
#define SWZ_XOR(v, m) \
  __int_as_float(__builtin_amdgcn_ds_swizzle(__float_as_int(v), (((m) << 10) | 0x1F)))

// floor(log2|v|)+1 ; exact for denormals, returns 0 for v==0 (harmless: group all-zero)
__device__ __forceinline__ int fexp(float v) {
#if __has_builtin(__builtin_amdgcn_frexp_expf)
  return __builtin_amdgcn_frexp_expf(v);       // v = m*2^e, m in [0.5,1)
#else
  return (int)((__float_as_uint(v) >> 23) & 0xFFu) - 126;
#endif
}

// e4m3 fake-quantize one element given the group's shared exponent `se`.
// All scalings are exact powers of two (v_ldexp_f32); rounding is
// half-away-from-zero via floor(|m|+0.5); clip at +-448 == mm<=14 when pe==8.
// Quant+dequant fused into one ldexp: exact, since |q| >= 2^-6 is never denormal.
__device__ __forceinline__ float mxq_elem(float x, int se) {
  float xs = ldexpf(x, -se);                   // x / 2^shared_exp   (exact)
  int   pe = fexp(xs) - 1;                     // floor(log2|xs|)    (exact)
  pe = pe < -6 ? -6 : pe;                      // e4m3 min exponent (denorms below)
  float m  = ldexpf(xs, 3 - pe);               // mantissa domain, |m| in [8,16)
  float mm = floorf(fabsf(m) + 0.5f);          // round half away from zero
  mm = fminf(mm, (pe >= 8) ? 14.0f : 3.0e38f); // saturate: 448 = 14 * 2^5
  return copysignf(ldexpf(mm, (pe - 3) + se), xs); // q * scale, fused & exact
}

__global__ __launch_bounds__(256)
void mx_fq_kernel(const float* __restrict__ in, float* __restrict__ out,
                  int n4, unsigned stride4, int trips) {
  __shared__ v4f smem[WAVES_PER_BLOCK * DEPTH * 32];

  const int tid  = threadIdx.x;
  const int wave = tid >> 5;                   // wave32
  const int lane = tid & 31;
  const unsigned gid = blockIdx.x * blockDim.x + tid;

  // Escape the LDS array into the asm world so the async-to-LDS writes
  // (memory clobber) are known to alias it; keeps the ds_load reads honest.
  asm volatile("" :: "v"((void*)smem) : "memory");

  // Static LDS allocation starts at LDS offset 0: raw byte address of
  // smem[(wave*DEPTH + slot)*32 + lane] is just the element offset * 16.
  const unsigned lds0 = (unsigned)((wave * DEPTH) * 32 + lane) * 16u;
  const unsigned long long sbase = (unsigned long long)in;

  auto issue = [&](int t, int slot) {
    unsigned idx = gid + (unsigned)t * stride4;
    if (idx >= (unsigned)n4) idx = (unsigned)n4 - 1u;  // clamped dummy prefetch
    unsigned voff  = idx << 4;                         // byte offset (16B/elem)
    unsigned laddr = lds0 + (unsigned)slot * (32u * 16u);
    // GVS mode: reads global address sbase + voff, writes to the LDS address in laddr. ASYNCcnt-tracked.
    // NT hint: 512 MiB single-pass stream, keep it out of WGP$/L2 working set.
    asm volatile("global_load_async_to_lds_b128 %0, %1, %2 th:TH_LOAD_NT"
                 :: "v"(laddr), "v"(voff), "s"(sbase)
                 : "memory");
  };

#pragma unroll
  for (int s = 0; s < DEPTH; ++s) issue(s, s);

  v4f* __restrict__ vout = (v4f*)out;

  for (int t = 0; t < trips; ++t) {
    // DEPTH outstanding -> waiting to <=DEPTH-1 guarantees oldest tile landed
    // (async loads complete in order per ISA 08 §4.1)
    asm volatile("s_wait_asynccnt 3" ::: "memory");

    const int slot = t & (DEPTH - 1);
    v4f v = smem[(wave * DEPTH + slot) * 32 + lane];

    // MX group of 32 = 8 lanes x float4: in-lane amax, then 3 xor-swaps
    float a = fmaxf(fmaxf(fabsf(v.x), fabsf(v.y)),
                    fmaxf(fabsf(v.z), fabsf(v.w)));
    a = fmaxf(a, SWZ_XOR(a, 1));
    a = fmaxf(a, SWZ_XOR(a, 2));
    a = fmaxf(a, SWZ_XOR(a, 4));

    int se = fexp(a) - 9;                      // floor(log2 amax) - EMAX(8)
    se = se > 127 ? 128 : (se < -127 ? -127 : se);

    v4f r;
    r.x = mxq_elem(v.x, se);
    r.y = mxq_elem(v.y, se);
    r.z = mxq_elem(v.z, se);
    r.w = mxq_elem(v.w, se);

    unsigned idx = gid + (unsigned)t * stride4;
    if (idx < (unsigned)n4)
      __builtin_nontemporal_store(r, vout + idx);   // b128 store, th:NT

    issue(t + DEPTH, slot);                    // refill the slot just consumed
  }
}

extern "C" void kernel_launch(void* const* d_in, const int* in_sizes, int n_in,
                              void* d_out, int out_size, void* d_ws, size_t ws_size,
                              hipStream_t stream) {
  (void)n_in; (void)out_size; (void)d_ws; (void)ws_size;
  const float* x = (const float*)d_in[0];
  float* out = (float*)d_out;
  long long n = (long long)in_sizes[0];        // 8192*8192, divisible by 4
  int n4 = (int)(n >> 2);

  const int block = 256;                       // 8 wave32 waves
  long long want = ((long long)n4 + block - 1) / block;
  int blocks = (int)(want < 4096 ? want : 4096);  // ~1M threads, 16 trips each
  if (blocks < 1) blocks = 1;

  unsigned stride4 = (unsigned)blocks * (unsigned)block;
  int trips = (int)(((long long)n4 + stride4 - 1) / stride4);

  mx_fq_kernel<<<blocks, block, 0, stream>>>(x, out, n4, stride4, trips);
}